// SCNNModel_35785667510310
// MI455X (gfx1250) — compile-verified
//
#include <hip/hip_runtime.h>
#include <stdint.h>

// ---------------- sizes / workspace layout (floats) ----------------
#define N_SPKIN     49152      // 16*3*32*32
#define N_C1        524288     // 16*32*32*32
#define N_P1        131072     // 16*32*16*16
#define N_C2        262144     // 16*64*16*16
#define N_P2        65536      // 16*64*8*8
#define N_FC1       8192       // 16*512
#define N_FC2       160        // 16*10
#define N_FC1P      65536      // 8 chunks * 8192 split-K partials

#define OFF_SPKIN   0
#define OFF_MP0     (OFF_SPKIN + N_SPKIN)
#define OFF_SPK1    (OFF_MP0   + N_C1)
#define OFF_MP1     (OFF_SPK1  + N_C1)
#define OFF_SPK2    (OFF_MP1   + N_P1)
#define OFF_MP2     (OFF_SPK2  + N_P1)
#define OFF_SPK3    (OFF_MP2   + N_C2)
#define OFF_MP3     (OFF_SPK3  + N_C2)
#define OFF_SPK4    (OFF_MP3   + N_P2)
#define OFF_MP4     (OFF_SPK4  + N_P2)
#define OFF_SPK5    (OFF_MP4   + N_FC1)
#define OFF_MP5     (OFF_SPK5  + N_FC1)
#define OFF_LC5     (OFF_MP5   + N_FC2)
#define OFF_FC1P    (OFF_LC5   + N_FC2)
#define WS_FLOATS   (OFF_FC1P  + N_FC1P)   // ~2.10M floats ~= 8.4 MB

#define THR    1.0f
#define DECAY  0.99f
#define PTHR   0.75f
#define STEPS  100

typedef float v2f __attribute__((ext_vector_type(2)));
typedef float v8f __attribute__((ext_vector_type(8)));

#define WMMA_F32(A, B, C) \
  __builtin_amdgcn_wmma_f32_16x16x4_f32(false, (A), false, (B), (short)0, (C), false, false)

// ---------------- threefry2x32 (JAX-compatible, 20 rounds) ----------------
__host__ __device__ inline void threefry2x32(uint32_t k0, uint32_t k1,
                                             uint32_t& x0, uint32_t& x1) {
  const uint32_t ks0 = k0, ks1 = k1, ks2 = 0x1BD11BDAu ^ k0 ^ k1;
  x0 += ks0; x1 += ks1;
#define TFR(r) { x0 += x1; x1 = (x1 << (r)) | (x1 >> (32 - (r))); x1 ^= x0; }
  TFR(13) TFR(15) TFR(26) TFR(6)
  x0 += ks1; x1 += ks2 + 1u;
  TFR(17) TFR(29) TFR(16) TFR(24)
  x0 += ks2; x1 += ks0 + 2u;
  TFR(13) TFR(15) TFR(26) TFR(6)
  x0 += ks0; x1 += ks1 + 3u;
  TFR(17) TFR(29) TFR(16) TFR(24)
  x0 += ks1; x1 += ks2 + 4u;
  TFR(13) TFR(15) TFR(26) TFR(6)
  x0 += ks2; x1 += ks0 + 5u;
#undef TFR
}

// ---------------- kernels ----------------
__global__ void k_init(float* ws, int n) {
  int i = blockIdx.x * blockDim.x + threadIdx.x;
  if (i < n) ws[i] = 0.0f;
}

// Poisson rate coding: out = (|x|/2 > U(key_t)) * sign(x)
__global__ void k_poisson(const float* __restrict__ x, float* __restrict__ spk,
                          uint32_t k0, uint32_t k1) {
  int i = blockIdx.x * blockDim.x + threadIdx.x;
  if (i >= N_SPKIN) return;
  const int half = N_SPKIN / 2;
  uint32_t c0, c1;
  bool first = i < half;
  if (first) { c0 = (uint32_t)i;          c1 = (uint32_t)(i + half); }
  else       { c0 = (uint32_t)(i - half); c1 = (uint32_t)i; }
  threefry2x32(k0, k1, c0, c1);
  uint32_t bits = first ? c0 : c1;
  float r = __uint_as_float((bits >> 9) | 0x3f800000u) - 1.0f;
  float xv = x[i];
  spk[i] = (fabsf(xv) * 0.5f > r) ? (xv > 0.0f ? 1.0f : -1.0f) : 0.0f;
}

// conv1 (3->32, 5x5, pad2) implicit GEMM on V_WMMA_F32_16X16X4_F32.
// K ordered (tap, ci), ci padded 3->4 (pad slot zeroed via khalf mask).
// Dual accumulators (kw parity) break the WMMA dependency chain.
__global__ void k_conv1_lif(const float* __restrict__ spkin,
                            const float* __restrict__ w1,
                            const float* __restrict__ b1,
                            float* __restrict__ mp0, float* __restrict__ spk1) {
  const int lane  = threadIdx.x & 31;
  const int tile  = blockIdx.x * (blockDim.x >> 5) + (threadIdx.x >> 5); // 0..2047
  const int m0    = (tile & 1) << 4;          // out-channel tile base
  const int n0    = (tile >> 1) << 4;         // pixel tile base (16-aligned)
  const int lh    = lane & 15;
  const int khalf = lane >> 4;

  const int p   = n0 + lh;
  const int pb  = p >> 10;
  const int rem = p & 1023;
  const int oh  = rem >> 5;   // wave-uniform
  const int ow  = rem & 31;

  const int m = m0 + lh;

  const float ayMask = khalf ? 0.0f : 1.0f;
  const float* wbase = w1 + m * 75 + khalf * 50;
  const float* sbase = spkin + (pb * 3 + khalf * 2) * 1024;
  const int byOff = khalf ? 0 : 1024;   // clamped ci for .y

  v8f acc0 = {}, acc1 = {};
  int tap = 0;
  for (int kh = 0; kh < 5; ++kh) {
    const int ih  = oh + kh - 2;
    const int ihc = min(max(ih, 0), 31);
    const bool hok = (unsigned)ih < 32u;
#pragma unroll
    for (int kw = 0; kw < 5; ++kw, ++tap) {
      const int iw  = ow + kw - 2;
      const int iwc = min(max(iw, 0), 31);
      const float mask = (hok && (unsigned)iw < 32u) ? 1.0f : 0.0f;
      v2f a, bb;
      a.x = wbase[tap];
      a.y = ayMask * wbase[tap + 25];
      const float* sp = sbase + ihc * 32 + iwc;
      bb.x = mask * sp[0];
      bb.y = (mask * ayMask) * sp[byOff];
      if (kw & 1) acc1 = WMMA_F32(a, bb, acc1);
      else        acc0 = WMMA_F32(a, bb, acc0);
    }
  }
  v8f acc = acc0 + acc1;
#pragma unroll
  for (int r = 0; r < 8; ++r) {
    int mm  = m0 + r + 8 * khalf;
    int idx = ((pb * 32 + mm) * 32 + oh) * 32 + ow;
    float mp = mp0[idx] + acc[r] + b1[mm];
    float ex = mp > THR ? mp : 0.0f;
    mp -= ex;
    spk1[idx] = ex > 0.0f ? 1.0f : 0.0f;
    mp0[idx] = DECAY * mp;
  }
}

// 2x2 avgpool + pool neuron (thr 0.75, no decay)
__global__ void k_pool1(const float* __restrict__ spk1, float* __restrict__ mp1,
                        float* __restrict__ spk2) {
  int idx = blockIdx.x * blockDim.x + threadIdx.x;
  if (idx >= N_P1) return;
  int ow = idx & 15, oh = (idx >> 4) & 15, c = (idx >> 8) & 31, b = idx >> 13;
  int base = ((b * 32 + c) * 32 + 2 * oh) * 32 + 2 * ow;
  float s = spk1[base] + spk1[base + 1] + spk1[base + 32] + spk1[base + 33];
  float mp = mp1[idx] + 0.25f * s;
  float ex = mp > PTHR ? mp : 0.0f;
  mp -= ex;
  spk2[idx] = ex > 0.0f ? 1.0f : 0.0f;
  mp1[idx] = mp;
}

// conv2 (32->64, 5x5, pad2) implicit GEMM, K ordered (tap, ci); dual accumulators.
__global__ void k_conv2_lif(const float* __restrict__ spk2,
                            const float* __restrict__ w2,
                            const float* __restrict__ b2,
                            float* __restrict__ mp2, float* __restrict__ spk3) {
  const int lane  = threadIdx.x & 31;
  const int tile  = blockIdx.x * (blockDim.x >> 5) + (threadIdx.x >> 5);
  const int m0    = (tile & 3) << 4;   // channel tile base
  const int n0    = (tile >> 2) << 4;  // pixel tile base
  const int lh    = lane & 15;
  const int khalf = lane >> 4;

  const int p    = n0 + lh;
  const int pb   = p >> 8;     // batch (wave-uniform)
  const int prem = p & 255;
  const int oh   = prem >> 4;  // wave-uniform
  const int ow   = lh;
  const int m    = m0 + lh;

  const float* aBase = w2 + m * 800 + khalf * 50;
  const float* bBase = spk2 + (pb * 32 + khalf * 2) * 256;

  v8f acc0 = {}, acc1 = {};
  int tap = 0;
  for (int kh = 0; kh < 5; ++kh) {
    const int ih  = oh + kh - 2;
    const int ihc = min(max(ih, 0), 15);
    const bool hok = (unsigned)ih < 16u;
    for (int kw = 0; kw < 5; ++kw, ++tap) {
      const int iw  = ow + kw - 2;
      const int iwc = min(max(iw, 0), 15);
      const float mask = (hok && (unsigned)iw < 16u) ? 1.0f : 0.0f;
      const float* ap = aBase + tap;
      const float* bp = bBase + ihc * 16 + iwc;
#pragma unroll
      for (int ci0 = 0; ci0 < 32; ci0 += 4) {
        v2f a, bb;
        a.x  = ap[ci0 * 25];
        a.y  = ap[ci0 * 25 + 25];
        bb.x = mask * bp[ci0 * 256];
        bb.y = mask * bp[ci0 * 256 + 256];
        if ((ci0 >> 2) & 1) acc1 = WMMA_F32(a, bb, acc1);
        else                acc0 = WMMA_F32(a, bb, acc0);
      }
    }
  }
  v8f acc = acc0 + acc1;
#pragma unroll
  for (int r = 0; r < 8; ++r) {
    int mm  = m0 + r + 8 * khalf;
    int idx = (pb * 64 + mm) * 256 + prem;
    float mp = mp2[idx] + acc[r] + b2[mm];
    float ex = mp > THR ? mp : 0.0f;
    mp -= ex;
    spk3[idx] = ex > 0.0f ? 1.0f : 0.0f;
    mp2[idx] = DECAY * mp;
  }
}

__global__ void k_pool2(const float* __restrict__ spk3, float* __restrict__ mp3,
                        float* __restrict__ spk4) {
  int idx = blockIdx.x * blockDim.x + threadIdx.x;
  if (idx >= N_P2) return;
  int ow = idx & 7, oh = (idx >> 3) & 7, c = (idx >> 6) & 63, b = idx >> 12;
  int base = ((b * 64 + c) * 16 + 2 * oh) * 16 + 2 * ow;
  float s = spk3[base] + spk3[base + 1] + spk3[base + 16] + spk3[base + 17];
  float mp = mp3[idx] + 0.25f * s;
  float ex = mp > PTHR ? mp : 0.0f;
  mp -= ex;
  spk4[idx] = ex > 0.0f ? 1.0f : 0.0f;   // flat layout == flatten(B,4096)
  mp3[idx] = mp;
}

// fc1 split-K partials: 8 K-chunks x 32 N-tiles = 256 waves, 128 WMMA each
// (dual accumulators, contiguous b64 fragment loads, wf1 stream prefetch).
__global__ void k_fc1_partial(const float* __restrict__ spk4,
                              const float* __restrict__ wf1,
                              float* __restrict__ part) {
  const int lane  = threadIdx.x & 31;
  const int tile  = blockIdx.x * (blockDim.x >> 5) + (threadIdx.x >> 5); // 0..255
  const int n0    = (tile & 31) << 4;
  const int chunk = tile >> 5;            // 0..7 (512 K each)
  const int lh    = lane & 15;
  const int khalf = lane >> 4;
  const int brow  = lh;
  const int ncol  = n0 + lh;

  const v2f* ap = (const v2f*)(spk4 + brow * 4096) + chunk * 256;
  const v2f* bp = (const v2f*)(wf1 + (size_t)ncol * 4096) + chunk * 256;

  v8f acc0 = {}, acc1 = {};
  for (int kc = 0; kc < 256; kc += 32) {          // 64 K per chunk iteration
    __builtin_prefetch(&bp[kc + 32], 0, 0);       // wf1 stream ahead 256B
#pragma unroll
    for (int k2 = 0; k2 < 32; k2 += 4) {
      v2f a0 = ap[kc + k2 + khalf];
      v2f b0 = bp[kc + k2 + khalf];
      acc0 = WMMA_F32(a0, b0, acc0);
      v2f a1 = ap[kc + k2 + 2 + khalf];
      v2f b1 = bp[kc + k2 + 2 + khalf];
      acc1 = WMMA_F32(a1, b1, acc1);
    }
  }
  v8f acc = acc0 + acc1;
#pragma unroll
  for (int r = 0; r < 8; ++r) {
    int bbatch = r + 8 * khalf;
    part[chunk * N_FC1 + bbatch * 512 + ncol] = acc[r];
  }
}

// fc1 reduce (fixed order => deterministic) + bias + LIF
__global__ void k_fc1_reduce(const float* __restrict__ part,
                             const float* __restrict__ bf1,
                             float* __restrict__ mp4, float* __restrict__ spk5) {
  int i = blockIdx.x * blockDim.x + threadIdx.x;
  if (i >= N_FC1) return;
  int n = i & 511;
  float s = bf1[n];
#pragma unroll
  for (int c = 0; c < 8; ++c) s += part[c * N_FC1 + i];
  float mp = mp4[i] + s;
  float ex = mp > THR ? mp : 0.0f;
  mp -= ex;
  spk5[i] = ex > 0.0f ? 1.0f : 0.0f;
  mp4[i] = DECAY * mp;
}

// fc2: one wave per output element; lane-parallel dot + shuffle butterfly
// reduce (fixed order, deterministic), then membrane + leaky accumulation.
__global__ void k_fc2(const float* __restrict__ spk5,
                      const float* __restrict__ wf2,
                      const float* __restrict__ bf2,
                      float* __restrict__ mp5, float* __restrict__ lc5) {
  const int wave = blockIdx.x * (blockDim.x >> 5) + (threadIdx.x >> 5); // 0..159
  const int lane = threadIdx.x & 31;
  const int b = wave / 10, j = wave - b * 10;
  const float* s = spk5 + b * 512;
  const float* w = wf2 + j * 512;
  float v = 0.0f;
#pragma unroll
  for (int k = 0; k < 512; k += 32) v += s[k + lane] * w[k + lane];
#pragma unroll
  for (int off = 16; off > 0; off >>= 1) v += __shfl_xor(v, off, 32);
  if (lane == 0) {
    float dot = v + bf2[j];
    float mp = mp5[wave] + dot;
    lc5[wave] = DECAY * (lc5[wave] + dot);
    float ex = mp > THR ? mp : 0.0f;
    mp -= ex;
    mp5[wave] = DECAY * mp;
  }
}

__global__ void k_final(const float* __restrict__ lc5, float* __restrict__ out) {
  int i = blockIdx.x * blockDim.x + threadIdx.x;
  if (i < N_FC2) out[i] = lc5[i] * (1.0f / (float)STEPS);
}

// ---------------- host driver ----------------
extern "C" void kernel_launch(void* const* d_in, const int* in_sizes, int n_in,
                              void* d_out, int out_size, void* d_ws, size_t ws_size,
                              hipStream_t stream) {
  (void)in_sizes; (void)n_in; (void)out_size; (void)ws_size;
  const float* x   = (const float*)d_in[0];
  const float* w1  = (const float*)d_in[1];
  const float* b1  = (const float*)d_in[2];
  const float* w2  = (const float*)d_in[3];
  const float* b2  = (const float*)d_in[4];
  const float* wf1 = (const float*)d_in[5];
  const float* bf1 = (const float*)d_in[6];
  const float* wf2 = (const float*)d_in[7];
  const float* bf2 = (const float*)d_in[8];
  float* ws  = (float*)d_ws;
  float* out = (float*)d_out;

  float* spkin = ws + OFF_SPKIN;
  float* mp0   = ws + OFF_MP0;   float* spk1 = ws + OFF_SPK1;
  float* mp1   = ws + OFF_MP1;   float* spk2 = ws + OFF_SPK2;
  float* mp2   = ws + OFF_MP2;   float* spk3 = ws + OFF_SPK3;
  float* mp3   = ws + OFF_MP3;   float* spk4 = ws + OFF_SPK4;
  float* mp4   = ws + OFF_MP4;   float* spk5 = ws + OFF_SPK5;
  float* mp5   = ws + OFF_MP5;   float* lc5  = ws + OFF_LC5;
  float* fc1p  = ws + OFF_FC1P;

  k_init<<<(WS_FLOATS + 255) / 256, 256, 0, stream>>>(ws, WS_FLOATS);

  for (int t = 0; t < STEPS; ++t) {
    // key = fold_in(key(42), t) computed on host (JAX threefry)
    uint32_t fk0 = 0u, fk1 = (uint32_t)t;
    threefry2x32(0u, 42u, fk0, fk1);

    k_poisson<<<(N_SPKIN + 255) / 256, 256, 0, stream>>>(x, spkin, fk0, fk1);
    k_conv1_lif<<<512, 128, 0, stream>>>(spkin, w1, b1, mp0, spk1);     // 2048 waves
    k_pool1<<<N_P1 / 256, 256, 0, stream>>>(spk1, mp1, spk2);
    k_conv2_lif<<<256, 128, 0, stream>>>(spk2, w2, b2, mp2, spk3);      // 1024 waves
    k_pool2<<<N_P2 / 256, 256, 0, stream>>>(spk3, mp3, spk4);
    k_fc1_partial<<<64, 128, 0, stream>>>(spk4, wf1, fc1p);             // 256 waves
    k_fc1_reduce<<<N_FC1 / 256, 256, 0, stream>>>(fc1p, bf1, mp4, spk5);
    k_fc2<<<40, 128, 0, stream>>>(spk5, wf2, bf2, mp5, lc5);            // 160 waves
  }
  k_final<<<1, 192, 0, stream>>>(lc5, out);
}